// RoIPoolingLayer_1546188226965
// MI455X (gfx1250) — compile-verified
//
#include <hip/hip_runtime.h>
#include <hip/hip_bf16.h>

namespace {
constexpr int kP = 7;     // pooled grid
constexpr int kH = 50;
constexpr int kW = 50;
constexpr int kC = 512;   // channels
constexpr int kB = 2;
constexpr int kR = 64;    // rois per batch
constexpr int kThreads = 128;              // 4 waves of 32; thread owns 4 channels
constexpr int kMaxPix  = 16;               // max bin footprint = 4x4 pixels
constexpr size_t kLdsBytes = (size_t)kMaxPix * kThreads * 16;  // 32 KB
}

// One workgroup per output bin (b, r, py, px). Thread t handles channels
// [4t, 4t+4). For each pixel of the bin we issue an async global->LDS B128
// copy (ASYNCcnt-tracked, no VGPR destination), wait once, then max-reduce
// from LDS. Each thread consumes only the LDS bytes it copied -> no barrier.
// Grid is (49, R, B) so the (b, r) decode comes free from blockIdx.y/z.
__global__ __launch_bounds__(kThreads) void roipool7x7_async(
    const float* __restrict__ xmaps,   // (B, H, W, C) f32
    const int*   __restrict__ xrois,   // (B, R, 4)   i32: y0, x0, hh, ww
    float*       __restrict__ out)     // (B, R, P, P, C) f32
{
    extern __shared__ float tile[];    // dynamic LDS only => raw LDS offset 0

    const int qq = blockIdx.x;         // py*7 + px, 0..48
    const int py = qq / kP;
    const int px = qq - py * kP;
    const int r  = blockIdx.y;
    const int b  = blockIdx.z;

    const int* roi = xrois + ((size_t)b * kR + r) * 4;
    const int y0 = roi[0], x0 = roi[1], hh = roi[2], ww = roi[3];

    // Bin bounds, matching the reference's float32 arithmetic exactly:
    // step = extent/7 (f32); start = trunc(i*step); end = trunc((i+1)*step),
    // last bin end forced to extent; size = max(end-start, 1).
    const float ystep = (float)hh / 7.0f;
    const float xstep = (float)ww / 7.0f;
    const int ys = (int)((float)py * ystep);
    const int ye = (py == kP - 1) ? hh : (int)((float)(py + 1) * ystep);
    const int xs = (int)((float)px * xstep);
    const int xe = (px == kP - 1) ? ww : (int)((float)(px + 1) * xstep);
    int sy = max(ye - ys, 1);
    int sx = max(xe - xs, 1);
    sy = min(sy, 4);                   // hh,ww <= 25 => bin extent <= ceil(25/7)=4
    sx = min(sx, 4);

    const unsigned cByte = (unsigned)threadIdx.x * 16u;       // this thread's 4 channels
    const char* base = (const char*)xmaps + (size_t)b * kH * kW * kC * sizeof(float);

    const int np = sy * sx;
    // Issue all async bulk copies up-front: pixel p goes to 16-byte LDS slot p*128 + tid.
    for (int p = 0; p < np; ++p) {
        const int dy = p / sx;
        const int dx = p - dy * sx;
        const int y  = min(y0 + ys + dy, kH - 1);             // never clips for valid bins
        const int x  = min(x0 + xs + dx, kW - 1);
        const char* g = base + (size_t)(y * kW + x) * kC * sizeof(float) + cByte;
        const unsigned lds = (unsigned)(p * kThreads) * 16u + cByte;
        asm volatile("global_load_async_to_lds_b128 %0, %1, off"
                     :
                     : "v"(lds), "v"(g)
                     : "memory");
    }
    // Drain the async DMA before touching LDS (per-wave; no cross-thread sharing).
    asm volatile("s_wait_asynccnt 0x0" ::: "memory");

    const float4* lt = (const float4*)tile;
    float4 m = lt[threadIdx.x];                               // p = 0 always exists
    for (int p = 1; p < np; ++p) {
        const float4 v = lt[(size_t)p * kThreads + threadIdx.x];
        m.x = fmaxf(m.x, v.x);
        m.y = fmaxf(m.y, v.y);
        m.z = fmaxf(m.z, v.z);
        m.w = fmaxf(m.w, v.w);
    }

    const size_t bin = (((size_t)b * kR + r) * (kP * kP)) + qq;  // output bin index
    float4* o = (float4*)((char*)out + bin * kC * sizeof(float) + cByte);
    *o = m;
}

extern "C" void kernel_launch(void* const* d_in, const int* in_sizes, int n_in,
                              void* d_out, int out_size, void* d_ws, size_t ws_size,
                              hipStream_t stream) {
    const float* xmaps = (const float*)d_in[0];   // (2, 50, 50, 512) f32
    const int*   xrois = (const int*)d_in[1];     // (2, 64, 4) i32
    float*       out   = (float*)d_out;           // (2, 64, 7, 7, 512) f32

    roipool7x7_async<<<dim3(kP * kP, kR, kB), dim3(kThreads), (unsigned)kLdsBytes, stream>>>(
        xmaps, xrois, out);
}